// Block_71562745085971
// MI455X (gfx1250) — compile-verified
//
#include <hip/hip_runtime.h>
#include <hip/hip_bf16.h>
#include <math.h>

#define B_   8
#define D_   4096
#define H_   32
#define DH_  128
#define DFF_ 16384
#define L_   4096
#define EPS_ 1e-5f

typedef __attribute__((ext_vector_type(2))) float v2f;
typedef __attribute__((ext_vector_type(8))) float v8f;

__device__ __forceinline__ float4 ld4(const float* p) { return *(const float4*)p; }

__device__ __forceinline__ float waveSum(float v) {
#pragma unroll
  for (int o = 16; o > 0; o >>= 1) v += __shfl_xor(v, o, 32);
  return v;
}
__device__ __forceinline__ float waveMax(float v) {
#pragma unroll
  for (int o = 16; o > 0; o >>= 1) v = fmaxf(v, __shfl_xor(v, o, 32));
  return v;
}
// blockDim.x == 256 (8 waves). red must be >= 8 floats, distinct per call site.
__device__ __forceinline__ float blockSum(float v, float* red) {
  const int lane = threadIdx.x & 31, w = threadIdx.x >> 5;
  v = waveSum(v);
  if (lane == 0) red[w] = v;
  __syncthreads();
  if (w == 0) {
    float t = (lane < 8) ? red[lane] : 0.0f;
    t = waveSum(t);
    if (lane == 0) red[0] = t;
  }
  __syncthreads();
  return red[0];
}
__device__ __forceinline__ float blockMax(float v, float* red) {
  const int lane = threadIdx.x & 31, w = threadIdx.x >> 5;
  v = waveMax(v);
  if (lane == 0) red[w] = v;
  __syncthreads();
  if (w == 0) {
    float t = (lane < 8) ? red[lane] : -3.402823466e38f;
    t = waveMax(t);
    if (lane == 0) red[0] = t;
  }
  __syncthreads();
  return red[0];
}

// ---------------- LayerNorm: one block per row of [rows, D_] ----------------
__global__ void __launch_bounds__(256) ln_kernel(const float* __restrict__ x,
                                                 const float* __restrict__ g,
                                                 const float* __restrict__ b,
                                                 float* __restrict__ out) {
  __shared__ float redA[8], redB[8];
  const int row = blockIdx.x;
  const float* xr = x + (size_t)row * D_;
  float s = 0.f, ss = 0.f;
  for (int i = threadIdx.x; i < D_; i += 256) {
    float v = xr[i]; s += v; ss += v * v;
  }
  s  = blockSum(s, redA);
  ss = blockSum(ss, redB);
  const float mean = s * (1.0f / D_);
  const float var  = ss * (1.0f / D_) - mean * mean;
  const float rstd = rsqrtf(var + EPS_);
  float* orow = out + (size_t)row * D_;
  for (int i = threadIdx.x; i < D_; i += 256)
    orow[i] = (xr[i] - mean) * rstd * g[i] + b[i];
}

// ---- WMMA fp32 GEMV: out[m,n] = sum_k A[m,k] * W[n,k], m in [0,8), split-K ----
// One wave computes one 16-column tile over one K-chunk. partial[chunk][m][n].
// tiles = N/16 must be a power of two: tile = wave & tileMask, chunk = wave >> tileShift.
__global__ void __launch_bounds__(256) gemv_wmma_kernel(
    const float* __restrict__ A, const float* __restrict__ W,
    float* __restrict__ partial, int K, int N, int tileShift, int kchunk) {
  const int wave  = blockIdx.x * 8 + (threadIdx.x >> 5);
  const int tile  = wave & ((1 << tileShift) - 1);
  const int chunk = wave >> tileShift;
  const int kbeg  = chunk * kchunk;
  const int kend  = kbeg + kchunk;

  const int lane = threadIdx.x & 31;
  const int half = lane >> 4;    // lo lanes: K-slots {0,1} ; hi lanes: {2,3}
  const int r    = lane & 15;    // A: row m ; B: col n (within tile)
  const bool avalid = (r < 8);   // rows 8..15 are zero padding

  const float* __restrict__ arow = A + (size_t)(r & 7) * K;       // aliased, zeroed below
  const float* __restrict__ wrow = W + (size_t)(tile * 16 + r) * K;

  v8f acc = {};
  // 32 K-values per iteration -> 8 WMMAs, 8 x b128 loads (branchless body)
#define GEMV_STEP(KOFF)                                                            \
  {                                                                                \
    const int ka = k + (KOFF) + half * 8;                                          \
    float4 a0 = ld4(arow + ka);                                                    \
    float4 a1 = ld4(arow + ka + 4);                                                \
    const float4 b0 = ld4(wrow + ka);                                              \
    const float4 b1 = ld4(wrow + ka + 4);                                          \
    a0.x = avalid ? a0.x : 0.f; a0.y = avalid ? a0.y : 0.f;                        \
    a0.z = avalid ? a0.z : 0.f; a0.w = avalid ? a0.w : 0.f;                        \
    a1.x = avalid ? a1.x : 0.f; a1.y = avalid ? a1.y : 0.f;                        \
    a1.z = avalid ? a1.z : 0.f; a1.w = avalid ? a1.w : 0.f;                        \
    v2f a, b;                                                                      \
    a[0] = a0.x; a[1] = a0.y; b[0] = b0.x; b[1] = b0.y;                            \
    acc = __builtin_amdgcn_wmma_f32_16x16x4_f32(false, a, false, b, (short)0, acc, false, false); \
    a[0] = a0.z; a[1] = a0.w; b[0] = b0.z; b[1] = b0.w;                            \
    acc = __builtin_amdgcn_wmma_f32_16x16x4_f32(false, a, false, b, (short)0, acc, false, false); \
    a[0] = a1.x; a[1] = a1.y; b[0] = b1.x; b[1] = b1.y;                            \
    acc = __builtin_amdgcn_wmma_f32_16x16x4_f32(false, a, false, b, (short)0, acc, false, false); \
    a[0] = a1.z; a[1] = a1.w; b[0] = b1.z; b[1] = b1.w;                            \
    acc = __builtin_amdgcn_wmma_f32_16x16x4_f32(false, a, false, b, (short)0, acc, false, false); \
  }

  for (int k = kbeg; k < kend; k += 32) {
    __builtin_prefetch(wrow + k + 512, 0, 1);  // global_prefetch on weight stream
    GEMV_STEP(0)
    GEMV_STEP(16)
  }
#undef GEMV_STEP

  // C/D layout: VGPR rr -> M=rr (lanes 0-15) / M=8+rr (lanes 16-31); N = lane&15.
  // Only lanes 0-15 hold real rows (m = rr in 0..7).
  if (lane < 16) {
    float* p = partial + (size_t)chunk * (B_ * (size_t)N) + tile * 16 + lane;
#pragma unroll
    for (int rr = 0; rr < 8; ++rr) p[(size_t)rr * N] = acc[rr];
  }
}

// ------- split-K reduce + epilogue: dst = act(sum_c partial + bias) + res -------
__global__ void __launch_bounds__(256) reduce_ep_kernel(
    const float* __restrict__ partial, const float* __restrict__ bias,
    const float* __restrict__ res, float* __restrict__ dst,
    int N, int chunks, int relu) {
  const int i = blockIdx.x * 256 + threadIdx.x;
  const int total = B_ * N;
  if (i >= total) return;
  float s = 0.f;
  for (int c = 0; c < chunks; ++c) s += partial[(size_t)c * total + i];
  if (bias) s += bias[i & (N - 1)];
  if (relu) s = fmaxf(s, 0.f);
  if (res) s += res[i];
  dst[i] = s;
}

// ------------- attention: one block per (b,h); wave-per-key-row -------------
__global__ void __launch_bounds__(256) attn_kernel(
    const float* __restrict__ kc, const float* __restrict__ vc,
    const float* __restrict__ q, const float* __restrict__ knew,
    const float* __restrict__ vnew, float* __restrict__ o) {
  __shared__ float sc[L_ + 8];
  __shared__ float redA[8], redB[8];
  __shared__ float obuf[8][DH_];

  const int bh = blockIdx.x;
  const int b  = bh >> 5;          // H_ == 32
  const int h  = bh & (H_ - 1);
  const int tid  = threadIdx.x;
  const int lane = tid & 31;
  const int wave = tid >> 5;
  const float scale = 0.08838834764831845f;  // 1/sqrt(128)

  const size_t headoff = (size_t)b * D_ + (size_t)h * DH_;
  const float4 q4 = ld4(q + headoff + lane * 4);

  // pass 1: scores (each wave owns whole key rows -> coalesced 512B row loads)
  const float* kbase = kc + (size_t)bh * L_ * DH_;
  const float* krown = knew + headoff;
  for (int key = wave; key <= L_; key += 8) {
    const float* kr = (key < L_) ? (kbase + (size_t)key * DH_) : krown;
    const float4 k4 = ld4(kr + lane * 4);
    float d = q4.x * k4.x + q4.y * k4.y + q4.z * k4.z + q4.w * k4.w;
    d = waveSum(d);
    if (lane == 0) sc[key] = d * scale;
  }
  __syncthreads();

  // softmax over L_+1 scores
  float m = -3.402823466e38f;
  for (int i = tid; i <= L_; i += 256) m = fmaxf(m, sc[i]);
  m = blockMax(m, redA);
  float s = 0.f;
  for (int i = tid; i <= L_; i += 256) {
    const float e = __expf(sc[i] - m);
    sc[i] = e;
    s += e;
  }
  s = blockSum(s, redB);
  const float inv = 1.0f / s;
  __syncthreads();

  // pass 2: o = sum_k p_k * V[k]
  const float* vbase = vc + (size_t)bh * L_ * DH_;
  const float* vrown = vnew + headoff;
  float4 acc = make_float4(0.f, 0.f, 0.f, 0.f);
  for (int key = wave; key <= L_; key += 8) {
    const float* vr = (key < L_) ? (vbase + (size_t)key * DH_) : vrown;
    const float4 v4 = ld4(vr + lane * 4);
    const float p = sc[key];
    acc.x += p * v4.x; acc.y += p * v4.y; acc.z += p * v4.z; acc.w += p * v4.w;
  }
  obuf[wave][lane * 4 + 0] = acc.x;
  obuf[wave][lane * 4 + 1] = acc.y;
  obuf[wave][lane * 4 + 2] = acc.z;
  obuf[wave][lane * 4 + 3] = acc.w;
  __syncthreads();
  if (tid < DH_) {
    float v = 0.f;
#pragma unroll
    for (int w = 0; w < 8; ++w) v += obuf[w][tid];
    o[headoff + tid] = v * inv;
  }
}

extern "C" void kernel_launch(void* const* d_in, const int* in_sizes, int n_in,
                              void* d_out, int out_size, void* d_ws, size_t ws_size,
                              hipStream_t stream) {
  const float* x    = (const float*)d_in[0];
  const float* kc   = (const float*)d_in[1];
  const float* vc   = (const float*)d_in[2];
  const float* Wq   = (const float*)d_in[3];
  const float* Wk   = (const float*)d_in[4];
  const float* Wv   = (const float*)d_in[5];
  const float* Wo   = (const float*)d_in[6];
  const float* g1   = (const float*)d_in[7];
  const float* b1   = (const float*)d_in[8];
  const float* g2   = (const float*)d_in[9];
  const float* b2   = (const float*)d_in[10];
  const float* Wup  = (const float*)d_in[11];
  const float* bup  = (const float*)d_in[12];
  const float* Wdn  = (const float*)d_in[13];
  const float* bdn  = (const float*)d_in[14];
  float* out = (float*)d_out;

  float* ws = (float*)d_ws;
  const size_t SZ = (size_t)B_ * D_;  // 32768
  float* h    = ws;
  float* qb   = ws + SZ;
  float* kn   = ws + 2 * SZ;
  float* vn   = ws + 3 * SZ;
  float* ob   = ws + 4 * SZ;
  float* x2   = ws + 5 * SZ;
  float* h2   = ws + 6 * SZ;
  float* ff   = ws + 7 * SZ;                       // B_*DFF_ = 131072
  float* part = ws + 7 * SZ + (size_t)B_ * DFF_;   // 262144 floats, reused

  // 1. h = LN1(x)
  ln_kernel<<<B_, 256, 0, stream>>>(x, g1, b1, h);

  // 2. q/k/v projections (WMMA, split-K=8 -> 2048 waves each)
  // D-out GEMVs: 256 tiles (shift 8), 8 K-chunks of 512 -> 2048 waves = 256 blocks
  const int gD = 256;
  const int rD = (B_ * D_) / 256;
  gemv_wmma_kernel<<<gD, 256, 0, stream>>>(h, Wq, part, D_, D_, 8, D_ / 8);
  reduce_ep_kernel<<<rD, 256, 0, stream>>>(part, nullptr, nullptr, qb, D_, 8, 0);
  gemv_wmma_kernel<<<gD, 256, 0, stream>>>(h, Wk, part, D_, D_, 8, D_ / 8);
  reduce_ep_kernel<<<rD, 256, 0, stream>>>(part, nullptr, nullptr, kn, D_, 8, 0);
  gemv_wmma_kernel<<<gD, 256, 0, stream>>>(h, Wv, part, D_, D_, 8, D_ / 8);
  reduce_ep_kernel<<<rD, 256, 0, stream>>>(part, nullptr, nullptr, vn, D_, 8, 0);

  // 3. attention over KV cache + new token
  attn_kernel<<<B_ * H_, 256, 0, stream>>>(kc, vc, qb, kn, vn, ob);

  // 4. x2 = x + o @ Wo^T
  gemv_wmma_kernel<<<gD, 256, 0, stream>>>(ob, Wo, part, D_, D_, 8, D_ / 8);
  reduce_ep_kernel<<<rD, 256, 0, stream>>>(part, nullptr, x, x2, D_, 8, 0);

  // 5. h2 = LN2(x2)
  ln_kernel<<<B_, 256, 0, stream>>>(x2, g2, b2, h2);

  // 6. ff = relu(h2 @ Wup^T + bup): 1024 tiles (shift 10), 2 K-chunks of 2048
  const int gU = 256;
  const int rU = (B_ * DFF_) / 256;
  gemv_wmma_kernel<<<gU, 256, 0, stream>>>(h2, Wup, part, D_, DFF_, 10, D_ / 2);
  reduce_ep_kernel<<<rU, 256, 0, stream>>>(part, bup, nullptr, ff, DFF_, 2, 1);

  // 7. out = x2 + ff @ Wdn^T + bdn: 256 tiles, 8 K-chunks of 2048
  gemv_wmma_kernel<<<gD, 256, 0, stream>>>(ff, Wdn, part, DFF_, D_, 8, DFF_ / 8);
  reduce_ep_kernel<<<rD, 256, 0, stream>>>(part, bdn, x2, out, D_, 8, 0);
}